// Linear_58643483460344
// MI455X (gfx1250) — compile-verified
//
#include <hip/hip_runtime.h>

typedef __attribute__((ext_vector_type(2))) float v2f;
typedef __attribute__((ext_vector_type(4))) float v4f;
typedef __attribute__((ext_vector_type(8))) float v8f;

#define BLOCK 32
#define NB 128
#define KDIM 4096
#define NDIM 4096
#define ROWS_PER_WG 256   // 8 wave32 waves x 32 rows each
#define THREADS 256
#define LSTRIDE 34        // padded col stride in LDS (bank-conflict avoidance)

// ---------------------------------------------------------------------------
// Build CSC lists for the block mask so the GEMM kernel accumulates each
// output tile in registers in a fixed (deterministic) order. nnz ~ 1638.
// ---------------------------------------------------------------------------
__global__ void bsmm_build_csc(const int* __restrict__ idx_i,
                               const int* __restrict__ idx_j, int nnz,
                               int* __restrict__ colptr,   // NB+1
                               int* __restrict__ colblk,   // nnz (block id)
                               int* __restrict__ coli)     // nnz (input block-col)
{
    if (threadIdx.x != 0 || blockIdx.x != 0) return;
    int cnt[NB];
    for (int j = 0; j < NB; ++j) cnt[j] = 0;
    for (int e = 0; e < nnz; ++e) cnt[idx_j[e]]++;
    int run = 0;
    for (int j = 0; j < NB; ++j) { colptr[j] = run; run += cnt[j]; cnt[j] = 0; }
    colptr[NB] = run;
    for (int e = 0; e < nnz; ++e) {
        int j = idx_j[e];
        int o = colptr[j] + cnt[j]++;
        colblk[o] = e;          // row-major nonzero order within each column
        coli[o]   = idx_i[e];
    }
}

// ---------------------------------------------------------------------------
// Block-sparse GEMM. One workgroup = one (256-row, 32-col) output tile.
// Each wave computes a 32x32 strip (2 M-tiles x 2 N-halves = 4 WMMA accums)
// with V_WMMA_F32_16X16X4_F32. w blocks staged through double-buffered LDS
// (column-major + pad -> each B fragment is one aligned ds_load_b64).
// ---------------------------------------------------------------------------
__global__ void __launch_bounds__(THREADS)
bsmm_wmma(const float* __restrict__ x, const float* __restrict__ w,
          const int* __restrict__ colptr, const int* __restrict__ colblk,
          const int* __restrict__ coli, float* __restrict__ y)
{
    __shared__ float wlds[2][BLOCK * LSTRIDE];   // 2 x 4.25KB double buffer

    const int j    = blockIdx.y;
    const int lane = threadIdx.x & 31;
    const int wave = threadIdx.x >> 5;
    const int half = lane >> 4;      // 0: lanes 0-15, 1: lanes 16-31
    const int lid  = lane & 15;
    const int m0   = blockIdx.x * ROWS_PER_WG + wave * 32;

    const int beg = colptr[j];
    const int cnt = colptr[j + 1] - beg;

    v8f c00 = {}, c01 = {};   // rows m0..m0+15,  N halves [0,16) / [16,32)
    v8f c10 = {}, c11 = {};   // rows m0+16..m0+31

    // A fragment bases: lane's rows; K phase per lane-half (16x4 f32 A layout)
    const float* xrow0 = x + (size_t)(m0 + lid) * KDIM + half * 2;
    const float* xrow1 = xrow0 + (size_t)16 * KDIM;

    // B fragment base offset in LDS (col-major, padded): col=lid(+16), k phase
    const int boff = lid * LSTRIDE + half * 2;

    // staging map: thread t covers row-major w floats [t*4, t*4+4)
    const int srow = threadIdx.x >> 3;        // k row  (0..31)
    const int scol = (threadIdx.x & 7) * 4;   // n col  (0..28 step 4)

    if (cnt > 0) {
        const float* wsrc =
            w + (size_t)colblk[beg] * (BLOCK * BLOCK) + srow * BLOCK + scol;
        v4f t0 = *(const v4f*)wsrc;
        wlds[0][(scol + 0) * LSTRIDE + srow] = t0[0];
        wlds[0][(scol + 1) * LSTRIDE + srow] = t0[1];
        wlds[0][(scol + 2) * LSTRIDE + srow] = t0[2];
        wlds[0][(scol + 3) * LSTRIDE + srow] = t0[3];
    }

    for (int t = 0; t < cnt; ++t) {
        __syncthreads();                       // buffer (t&1) is populated
        const int buf = t & 1;

        // prefetch next w block into registers while we compute (wave-uniform)
        const bool have_next = (t + 1) < cnt;
        v4f nxt = {};
        if (have_next) {
            const float* wsrc = w + (size_t)colblk[beg + t + 1] * (BLOCK * BLOCK)
                                  + srow * BLOCK + scol;
            nxt = *(const v4f*)wsrc;
        }

        const int    kbase = coli[beg + t] * BLOCK;
        const float* xa0   = xrow0 + kbase;
        const float* xa1   = xrow1 + kbase;
        const float* bl    = &wlds[buf][boff];

#pragma unroll
        for (int s = 0; s < 8; ++s) {
            v2f a0 = *(const v2f*)(xa0 + s * 4);                 // global b64
            v2f a1 = *(const v2f*)(xa1 + s * 4);                 // global b64
            v2f b0 = *(const v2f*)(bl + s * 4);                  // ds b64
            v2f b1 = *(const v2f*)(bl + 16 * LSTRIDE + s * 4);   // ds b64
            c00 = __builtin_amdgcn_wmma_f32_16x16x4_f32(
                      false, a0, false, b0, (short)0, c00, false, false);
            c01 = __builtin_amdgcn_wmma_f32_16x16x4_f32(
                      false, a0, false, b1, (short)0, c01, false, false);
            c10 = __builtin_amdgcn_wmma_f32_16x16x4_f32(
                      false, a1, false, b0, (short)0, c10, false, false);
            c11 = __builtin_amdgcn_wmma_f32_16x16x4_f32(
                      false, a1, false, b1, (short)0, c11, false, false);
        }

        if (have_next) {
            __syncthreads();                   // all waves done reading buf^1
            float* dst = &wlds[buf ^ 1][0];
            dst[(scol + 0) * LSTRIDE + srow] = nxt[0];
            dst[(scol + 1) * LSTRIDE + srow] = nxt[1];
            dst[(scol + 2) * LSTRIDE + srow] = nxt[2];
            dst[(scol + 3) * LSTRIDE + srow] = nxt[3];
        }
    }

    // Store D per the 16x16 f32 C/D layout: VGPR r -> row m0 + half*8 + r
    float* y0 = y + (size_t)(m0 + half * 8) * NDIM + j * BLOCK + lid;
    float* y1 = y0 + (size_t)16 * NDIM;
#pragma unroll
    for (int r = 0; r < 8; ++r) {
        y0[(size_t)r * NDIM]      = c00[r];
        y0[(size_t)r * NDIM + 16] = c01[r];
        y1[(size_t)r * NDIM]      = c10[r];
        y1[(size_t)r * NDIM + 16] = c11[r];
    }
}

// ---------------------------------------------------------------------------
extern "C" void kernel_launch(void* const* d_in, const int* in_sizes, int n_in,
                              void* d_out, int out_size, void* d_ws, size_t ws_size,
                              hipStream_t stream) {
    const float* x     = (const float*)d_in[0];
    const float* w     = (const float*)d_in[1];
    const int*   idx_i = (const int*)d_in[2];
    const int*   idx_j = (const int*)d_in[3];
    float*       y     = (float*)d_out;

    const int nnz = in_sizes[2];
    const int M   = in_sizes[0] / KDIM;

    int* colptr = (int*)d_ws;            // NB+1 ints
    int* colblk = colptr + (NB + 1);     // nnz ints
    int* coli   = colblk + nnz;          // nnz ints

    bsmm_build_csc<<<1, 32, 0, stream>>>(idx_i, idx_j, nnz, colptr, colblk, coli);

    dim3 grid(M / ROWS_PER_WG, NB);
    bsmm_wmma<<<grid, THREADS, 0, stream>>>(x, w, colptr, colblk, coli, y);
}